// SymmetricContraction_62345745268780
// MI455X (gfx1250) — compile-verified
//
#include <hip/hip_runtime.h>

// ---------------------------------------------------------------------------
// MACE symmetric contraction, MI455X (gfx1250, wave32, WMMA).
//
//   out[b,c] = sum_w v_w * ( S1[e,c,w]
//            + sum_x v_x * ( S2[e,c,w,x] + sum_i v_i * S3[e,c,w,x,i] ) )
//   with v = x[b,c,:], e = species(b),
//   S3[e,c,w,x,i] = sum_k U3[w,x,i,k] W3[e,k,c]   (hoisted per species)
//
// Main pass = 2.1M V_WMMA_F32_16X16X4_F32 ops (4.3 GFLOP f32).
// Round-3 fix: the t2 loop is macro-expanded so ALL register-array indices
// are literal constants -- the compiler's partial unroll had reintroduced
// v_cmp/v_cndmask select trees for xf[t2]. GRP=4 waves/block still share
// one staged S3[e,c] LDS block (4x less staging traffic).
// ---------------------------------------------------------------------------

#define B_N   4096
#define C_N   128
#define L_N   16
#define E_N   10
#define K3_N  23
#define K2_N  4
#define WX_N  256                  // L*L
#define GRP   4                    // tiles (waves) per block, share S3 block
#define MAXT  304                  // padded slots: <=266 tiles + <=30 pad, /4
#define NGRP  (MAXT / GRP)         // 76 groups

typedef float v2f __attribute__((ext_vector_type(2)));
typedef float v8f __attribute__((ext_vector_type(8)));

// ---------------- per-species table precompute ------------------------------

__global__ void sc_pre_s3(const float* __restrict__ U3,
                          const float* __restrict__ W3,
                          float* __restrict__ S3) {
  int t = blockIdx.x * blockDim.x + threadIdx.x;   // [0, E*C*WX*L)
  int i  = t & 15;
  int wx = (t >> 4) & 255;
  int c  = (t >> 12) & 127;
  int e  = t >> 19;
  const float* u = U3 + (wx * L_N + i) * K3_N;     // contiguous in k
  const float* w = W3 + e * K3_N * C_N + c;        // stride C over k
  float acc = 0.f;
#pragma unroll
  for (int k = 0; k < K3_N; ++k) acc += u[k] * w[k * C_N];
  S3[t] = acc;
}

__global__ void sc_pre_s2(const float* __restrict__ U2,
                          const float* __restrict__ W2,
                          float* __restrict__ S2) {
  int t = blockIdx.x * blockDim.x + threadIdx.x;   // [0, E*C*L*L)
  int x = t & 15;
  int w = (t >> 4) & 15;
  int c = (t >> 8) & 127;
  int e = t >> 15;
  const float* u  = U2 + (w * L_N + x) * K2_N;
  const float* wp = W2 + e * K2_N * C_N + c;
  float acc = 0.f;
#pragma unroll
  for (int k = 0; k < K2_N; ++k) acc += u[k] * wp[k * C_N];
  S2[t] = acc;
}

__global__ void sc_pre_s1(const float* __restrict__ U1,
                          const float* __restrict__ W1,
                          float* __restrict__ S1) {
  int t = blockIdx.x * blockDim.x + threadIdx.x;   // [0, E*C*L)
  int w = t & 15;
  int c = (t >> 4) & 127;
  int e = t >> 11;
  S1[t] = U1[w] * W1[e * C_N + c];                 // K1 == 1
}

// ---------------- species bucketing (single block) --------------------------
// Intra-bucket order is atomic-nondeterministic, but each node's result is
// independent and written exactly once -> d_out is deterministic.
// Tile slots are padded with -1 so each species starts at a GRP boundary:
// every group of GRP slots is species-uniform.

__global__ void sc_bucket(const float* __restrict__ y,
                          int* __restrict__ spec,
                          int* __restrict__ order,
                          int* __restrict__ tileS,
                          int* __restrict__ tileStart,
                          int* __restrict__ tileCnt) {
  __shared__ int cnt[E_N], cur[E_N];
  const int tid = threadIdx.x;                     // 256 threads
  if (tid < E_N) cnt[tid] = 0;
  __syncthreads();
  for (int b = tid; b < B_N; b += 256) {
    const float* yb = y + b * E_N;
    int e = 0;
#pragma unroll
    for (int q = 1; q < E_N; ++q)
      if (yb[q] > 0.5f) e = q;                     // one-hot argmax
    spec[b] = e;
    atomicAdd(&cnt[e], 1);
  }
  __syncthreads();
  if (tid == 0) {
    int acc = 0, slot = 0;
    for (int e = 0; e < E_N; ++e) {
      while (slot & (GRP - 1)) tileS[slot++] = -1; // align species to group
      cur[e] = acc;
      int n = cnt[e];
      for (int j = 0; j < n; j += 16) {
        tileS[slot]     = e;
        tileStart[slot] = acc + j;
        tileCnt[slot]   = (n - j < 16) ? (n - j) : 16;
        ++slot;
      }
      acc += n;
    }
    for (; slot < MAXT; ++slot) tileS[slot] = -1;
  }
  __syncthreads();
  for (int b = tid; b < B_N; b += 256) {
    int pos = atomicAdd(&cur[spec[b]], 1);
    order[pos] = b;
  }
}

// ---------------- main WMMA pass --------------------------------------------
// grid = (NGRP, C), block = 128 (4 waves). The 4 waves share one staged
// S3[e,c] LDS block (their tile-slots are species-uniform by construction);
// each wave runs 16 row-tiles x 4 K-chunks = 64 v_wmma_f32_16x16x4_f32.

// one K-chunk: A 16x4 fragment from LDS (lane->M row, half->K), chained WMMA
#define SC_K(T2, KK)                                                          \
  {                                                                           \
    v2f a = *(const v2f*)&sA[((T2)*16 + col) * 20 + ((KK)*4 + 2*hi)];         \
    acc = __builtin_amdgcn_wmma_f32_16x16x4_f32(                              \
        false, a, false, bf##KK, (short)0, acc, false, false);                \
  }

// one row tile (w == T2, a literal): S2-preloaded accumulator, 4 WMMAs,
// quadratic-stage dot over this lane's 8 rows, cubic weight xf[T2] (literal)
#define SC_TILE(T2)                                                           \
  {                                                                           \
    float4 s20 = *(const float4*)(s2p + (T2)*16 + 8*hi);                      \
    float4 s21 = *(const float4*)(s2p + (T2)*16 + 8*hi + 4);                  \
    v8f acc;                                                                  \
    acc[0] = s20.x; acc[1] = s20.y; acc[2] = s20.z; acc[3] = s20.w;           \
    acc[4] = s21.x; acc[5] = s21.y; acc[6] = s21.z; acc[7] = s21.w;           \
    SC_K(T2, 0) SC_K(T2, 1) SC_K(T2, 2) SC_K(T2, 3)                           \
    float p = acc[0]*xq0 + acc[1]*xq1 + acc[2]*xq2 + acc[3]*xq3               \
            + acc[4]*xq4 + acc[5]*xq5 + acc[6]*xq6 + acc[7]*xq7;              \
    rsum += p * xf[(T2)];                                                     \
  }

__global__ __launch_bounds__(32 * GRP)
void sc_main(const float* __restrict__ x,
             const float* __restrict__ S3,
             const float* __restrict__ S2,
             const float* __restrict__ S1,
             const int* __restrict__ order,
             const int* __restrict__ tileS,
             const int* __restrict__ tileStart,
             const int* __restrict__ tileCnt,
             float* __restrict__ out) {
  // S3[e][c] block, rows padded 16 -> 20 dwords: conflict-free ds reads,
  // 16B-aligned b128 stores. 256*20*4 = 20 KB of the 320 KB/WGP LDS.
  __shared__ float sA[WX_N * 20];

  const int g  = blockIdx.x;
  const int eg = tileS[g * GRP];                   // group species (-1 = empty)
  if (eg < 0) return;                              // block-uniform exit
  const int c    = blockIdx.y;
  const int tid  = threadIdx.x;                    // 0..127
  const int widx = tid >> 5;                       // wave in block
  const int lane = tid & 31;                       // wave32 lane
  const int col  = lane & 15;                      // node column (N)
  const int hi   = lane >> 4;                      // lane half -> K/M split

  const int slot = g * GRP + widx;
  const int myE  = tileS[slot];                    // eg or -1 (pad slot)

  // --- stage S3[eg][c] (16 KB) into LDS, 128 threads x 8 float4, coalesced --
  const float* s3p = S3 + (size_t)(eg * C_N + c) * (WX_N * L_N);
#pragma unroll
  for (int it = 0; it < 8; ++it) {
    int gi = it * 512 + tid * 4;                   // float index 0..4095
    float4 v = *(const float4*)(s3p + gi);
    int r = gi >> 4, i0 = gi & 15;                 // i0 in {0,4,8,12}
    *(float4*)&sA[r * 20 + i0] = v;                // 16B aligned
  }

  // --- per-lane node x-vector (lanes l and l+16 duplicate column col) ---
  int nid = -1;
  if (myE >= 0 && col < tileCnt[slot]) nid = order[tileStart[slot] + col];
  float xf[16];
  if (nid >= 0) {
    const float4* xp = (const float4*)(x + ((size_t)nid * C_N + c) * L_N);
    float4 v0 = xp[0], v1 = xp[1], v2 = xp[2], v3 = xp[3];
    xf[0]  = v0.x; xf[1]  = v0.y; xf[2]  = v0.z; xf[3]  = v0.w;
    xf[4]  = v1.x; xf[5]  = v1.y; xf[6]  = v1.z; xf[7]  = v1.w;
    xf[8]  = v2.x; xf[9]  = v2.y; xf[10] = v2.z; xf[11] = v2.w;
    xf[12] = v3.x; xf[13] = v3.y; xf[14] = v3.z; xf[15] = v3.w;
  } else {
#pragma unroll
    for (int q = 0; q < 16; ++q) xf[q] = 0.f;      // padded column -> zero
  }
  __syncthreads();                                 // staging visible to all

  if (myE < 0) return;                             // pad wave: no more barriers

  // --- B fragments (K=i, N=node), constant across all 16 row tiles.
  // 32-bit B 4x16: lane -> N, VGPR + lane-half -> K (mirrors A 16x4 table).
  // hi is 0/1: explicit 2-way selects only; all indices literal.
  const v2f bf0 = v2f{hi ? xf[2]  : xf[0],  hi ? xf[3]  : xf[1]};
  const v2f bf1 = v2f{hi ? xf[6]  : xf[4],  hi ? xf[7]  : xf[5]};
  const v2f bf2 = v2f{hi ? xf[10] : xf[8],  hi ? xf[11] : xf[9]};
  const v2f bf3 = v2f{hi ? xf[14] : xf[12], hi ? xf[15] : xf[13]};
  // quadratic-stage weights for this lane's 8 rows (rowj = j + 8*hi)
  const float xq0 = hi ? xf[8]  : xf[0];
  const float xq1 = hi ? xf[9]  : xf[1];
  const float xq2 = hi ? xf[10] : xf[2];
  const float xq3 = hi ? xf[11] : xf[3];
  const float xq4 = hi ? xf[12] : xf[4];
  const float xq5 = hi ? xf[13] : xf[5];
  const float xq6 = hi ? xf[14] : xf[6];
  const float xq7 = hi ? xf[15] : xf[7];

  const float* s2p = S2 + (size_t)(eg * C_N + c) * (L_N * L_N);
  float rsum = 0.f;

  // 16 row tiles, macro-expanded: every index is a literal constant.
  SC_TILE(0)  SC_TILE(1)  SC_TILE(2)  SC_TILE(3)
  SC_TILE(4)  SC_TILE(5)  SC_TILE(6)  SC_TILE(7)
  SC_TILE(8)  SC_TILE(9)  SC_TILE(10) SC_TILE(11)
  SC_TILE(12) SC_TILE(13) SC_TILE(14) SC_TILE(15)

  // combine the two row-halves of each column
  float rtot = rsum + __shfl_xor(rsum, 16, 32);

  if (hi == 0 && nid >= 0) {
    const float* s1p = S1 + (size_t)(eg * C_N + c) * L_N;
    float s = 0.f;
#pragma unroll
    for (int w = 0; w < L_N; ++w) s += s1p[w] * xf[w];   // linear stage
    out[(size_t)nid * C_N + c] = s + rtot;
  }
}

// ---------------- launcher ---------------------------------------------------

extern "C" void kernel_launch(void* const* d_in, const int* in_sizes, int n_in,
                              void* d_out, int out_size, void* d_ws, size_t ws_size,
                              hipStream_t stream) {
  const float* x  = (const float*)d_in[0];   // (B,C,L)
  const float* y  = (const float*)d_in[1];   // (B,E) one-hot
  const float* U3 = (const float*)d_in[2];   // (L,L,L,K3)
  const float* U2 = (const float*)d_in[3];   // (L,L,K2)
  const float* U1 = (const float*)d_in[4];   // (L,K1)
  const float* W3 = (const float*)d_in[5];   // (E,K3,C)
  const float* W2 = (const float*)d_in[6];   // (E,K2,C)
  const float* W1 = (const float*)d_in[7];   // (E,K1,C)
  float* out = (float*)d_out;                // (B,C)

  // workspace carve-up (~22.4 MB total)
  char* ws = (char*)d_ws;
  const size_t S3_ELEMS = (size_t)E_N * C_N * WX_N * L_N;   // 5,242,880
  const size_t S2_ELEMS = (size_t)E_N * C_N * L_N * L_N;    //   327,680
  const size_t S1_ELEMS = (size_t)E_N * C_N * L_N;          //    20,480
  float* S3 = (float*)ws;                 ws += S3_ELEMS * 4;
  float* S2 = (float*)ws;                 ws += S2_ELEMS * 4;
  float* S1 = (float*)ws;                 ws += S1_ELEMS * 4;
  int* spec      = (int*)ws;              ws += B_N * 4;
  int* order     = (int*)ws;              ws += B_N * 4;
  int* tileS     = (int*)ws;              ws += MAXT * 4;
  int* tileStart = (int*)ws;              ws += MAXT * 4;
  int* tileCnt   = (int*)ws;              ws += MAXT * 4;
  (void)in_sizes; (void)n_in; (void)out_size; (void)ws_size;

  sc_pre_s3<<<dim3((unsigned)(S3_ELEMS / 256)), dim3(256), 0, stream>>>(U3, W3, S3);
  sc_pre_s2<<<dim3((unsigned)(S2_ELEMS / 256)), dim3(256), 0, stream>>>(U2, W2, S2);
  sc_pre_s1<<<dim3((unsigned)(S1_ELEMS / 256)), dim3(256), 0, stream>>>(U1, W1, S1);
  sc_bucket<<<dim3(1), dim3(256), 0, stream>>>(y, spec, order, tileS, tileStart, tileCnt);
  sc_main<<<dim3(NGRP, C_N), dim3(32 * GRP), 0, stream>>>(x, S3, S2, S1, order,
                                                          tileS, tileStart, tileCnt, out);
}